// NSAMSA_44667659878880
// MI455X (gfx1250) — compile-verified
//
#include <hip/hip_runtime.h>
#include <hip/hip_bf16.h>
#include <math.h>

typedef float v2f __attribute__((ext_vector_type(2)));
typedef float v8f __attribute__((ext_vector_type(8)));

#define NMTOK 8192
#define DIM   256
#define NH    8
#define EDIM  32
#define NBALL 512
#define KDIM  256   // GEMM reduction dim

#if __has_builtin(__builtin_amdgcn_tensor_load_to_lds)
#define HAVE_TDM 1
#else
#define HAVE_TDM 0
#endif
#if __has_include(<hip/amd_detail/amd_gfx1250_TDM.h>)
#define TDM_6ARG 1
#else
#define TDM_6ARG 0
#endif

// ---------------- workspace layout (float elements) ----------------
#define XPE_OFF 0
#define Q_OFF   (NMTOK*DIM)
#define K_OFF   (2*NMTOK*DIM)
#define V_OFF   (3*NMTOK*DIM)
#define AO_OFF  (4*NMTOK*DIM)
#define KM_OFF  (5*NMTOK*DIM)
#define IDX_OFF (5*NMTOK*DIM + NH*NBALL*EDIM)

// ================= Kernel 1: ball-relative positional encode =================
__global__ __launch_bounds__(256) void pe_kernel(
    const float* __restrict__ x, const float* __restrict__ pos,
    const float* __restrict__ Wpe, const float* __restrict__ bpe,
    float* __restrict__ xpe) {
  int ball = blockIdx.x;
  int t = threadIdx.x;
  __shared__ float sp[3][16];
  if (t < 48) {
    int c = t / 16, m = t % 16;
    sp[c][m] = pos[(size_t)(ball * 16 + m) * 3 + c];
  }
  __syncthreads();
  float mean[3];
#pragma unroll
  for (int c = 0; c < 3; ++c) {
    float s = 0.f;
#pragma unroll
    for (int m = 0; m < 16; ++m) s += sp[c][m];
    mean[c] = s * (1.f / 16.f);
  }
  int m = t >> 4, dl = t & 15;
  float r0 = sp[0][m] - mean[0];
  float r1 = sp[1][m] - mean[1];
  float r2 = sp[2][m] - mean[2];
  size_t tok = (size_t)ball * 16 + m;
#pragma unroll
  for (int j = 0; j < 16; ++j) {
    int d = dl + j * 16;
    xpe[tok * DIM + d] = x[tok * DIM + d]
        + r0 * Wpe[d * 3 + 0] + r1 * Wpe[d * 3 + 1] + r2 * Wpe[d * 3 + 2]
        + bpe[d];
  }
}

// ================= TDM: DMA a 64x64 f32 tile (row-major, K contiguous) into LDS
// with hardware padding of 4 DWORDs every 64 DWORDs (-> LDS rows of 68 floats).
#if HAVE_TDM
__device__ __forceinline__ void tdm_load_tile64(unsigned lds_addr,
                                                const float* gptr,
                                                unsigned nrows) {
  typedef unsigned int u32x4 __attribute__((ext_vector_type(4)));
  typedef int i32x4 __attribute__((ext_vector_type(4)));
  typedef int i32x8 __attribute__((ext_vector_type(8)));
  unsigned long long ga = (unsigned long long)(uintptr_t)gptr;
  u32x4 g0;
  g0[0] = 1u;                                   // count=1, user descriptor
  g0[1] = lds_addr;                             // LDS byte address of tile
  g0[2] = (unsigned)ga;                         // global_addr[31:0]
  g0[3] = ((unsigned)(ga >> 32) & 0x01FFFFFFu)  // global_addr[56:32]
        | (2u << 30);                           // type=2 ("image")
  i32x8 g1;
  g1[0] = (int)((2u << 16)      // data_size = 4 bytes
              | (1u << 20)      // pad_enable
              | (5u << 22)      // pad_interval: 64 DWORDs
              | (3u << 25));    // pad_amount: 4 DWORDs
  g1[1] = (int)((KDIM & 0xFFFFu) << 16);                      // tensor_dim0 lo16
  g1[2] = (int)(((KDIM >> 16) & 0xFFFFu)                      // tensor_dim0 hi16
              | ((nrows & 0xFFFFu) << 16));                   // tensor_dim1 lo16
  g1[3] = (int)(((nrows >> 16) & 0xFFFFu)                     // tensor_dim1 hi16
              | (64u << 16));                                 // tile_dim0 = 64
  g1[4] = 64;                                                 // tile_dim1 = 64
  g1[5] = KDIM;                                               // dim0_stride lo32
  g1[6] = 0;
  g1[7] = 0;
  i32x4 z4 = {0, 0, 0, 0};
#if TDM_6ARG
  i32x8 z8 = {0, 0, 0, 0, 0, 0, 0, 0};
  __builtin_amdgcn_tensor_load_to_lds(g0, g1, z4, z4, z8, 0);
#else
  __builtin_amdgcn_tensor_load_to_lds(g0, g1, z4, z4, 0);
#endif
}
#endif

// ================= WMMA f32 GEMM:  C[M,N] = A[M,256] * B[N,256]^T =================
// Block: 256 thr = 8 waves; macro-tile 64(M) x 64(N); each wave: two 16x16 C tiles.
// EPI 0: scatter QKV (cols decode h/e/s) + b_qkv.  EPI 1: plain store + bias.
template <int EPI, int NCOLS>
__global__ __launch_bounds__(256) void gemm_wmma_f32(
    const float* __restrict__ A, const float* __restrict__ B,
    const float* __restrict__ bias,
    float* __restrict__ d0, float* __restrict__ d1, float* __restrict__ d2) {
  __shared__ float As[64][68];   // pad 4 -> conflict-free strided reads
  __shared__ float Bs[64][68];

  const int t = threadIdx.x;
  const int row0 = blockIdx.y * 64;
  const int col0 = blockIdx.x * 64;

  const int lane = t & 31, wave = t >> 5;
  const int l16 = lane & 15, hi = lane >> 4;
  const int koff = hi * 2;
  const int tm = wave & 3, tg = wave >> 2;   // wave -> (row tile, col pair)

  v8f acc0 = {}, acc1 = {};

  const float* arow  = &As[tm * 16 + l16][0];
  const float* brow0 = &Bs[tg * 32 + l16][0];
  const float* brow1 = &Bs[tg * 32 + 16 + l16][0];

  for (int k0 = 0; k0 < KDIM; k0 += 64) {
#if HAVE_TDM
    // --- Tensor Data Mover: one wave DMAs both 64x64 panels into LDS ---
    if (wave == 0) {
      tdm_load_tile64((unsigned)(uintptr_t)&As[0][0],
                      A + (size_t)row0 * KDIM + k0, NMTOK);
      tdm_load_tile64((unsigned)(uintptr_t)&Bs[0][0],
                      B + (size_t)col0 * KDIM + k0, (unsigned)NCOLS);
      __builtin_amdgcn_s_wait_tensorcnt(0);
    }
    __syncthreads();
#else
    // --- fallback: cooperative b128 staging ---
#pragma unroll
    for (int i = 0; i < 4; ++i) {
      int f = t + 256 * i;
      int row = f >> 4;
      int kq = (f & 15) << 2;
      float4 av = *(const float4*)&A[(size_t)(row0 + row) * KDIM + k0 + kq];
      *(float4*)&As[row][kq] = av;
      float4 bv = *(const float4*)&B[(size_t)(col0 + row) * KDIM + k0 + kq];
      *(float4*)&Bs[row][kq] = bv;
      if (k0 + 64 < KDIM) {
        __builtin_prefetch(&A[(size_t)(row0 + row) * KDIM + k0 + 64 + kq], 0, 1);
        __builtin_prefetch(&B[(size_t)(col0 + row) * KDIM + k0 + 64 + kq], 0, 1);
      }
    }
    __syncthreads();
#endif

    // --- 16 k-steps of V_WMMA_F32_16X16X4_F32, two N tiles per wave ---
#pragma unroll
    for (int ks = 0; ks < 64; ks += 4) {
      v2f a  = *(const v2f*)(arow + ks + koff);
      v2f b0 = *(const v2f*)(brow0 + ks + koff);
      v2f b1 = *(const v2f*)(brow1 + ks + koff);
      acc0 = __builtin_amdgcn_wmma_f32_16x16x4_f32(
          false, a, false, b0, (short)0, acc0, false, false);
      acc1 = __builtin_amdgcn_wmma_f32_16x16x4_f32(
          false, a, false, b1, (short)0, acc1, false, false);
    }
    __syncthreads();
  }

  // --- epilogue: VGPR r holds (M=r, lanes0-15) / (M=r+8, lanes16-31), N=l16 ---
#pragma unroll
  for (int r = 0; r < 8; ++r) {
    int Mloc = r + 8 * hi;
    int row = row0 + tm * 16 + Mloc;
    int colA = col0 + tg * 32 + l16;
    int colB = colA + 16;
    float vA = acc0[r] + bias[colA];
    float vB = acc1[r] + bias[colB];
    if (EPI == 1) {
      d0[(size_t)row * NCOLS + colA] = vA;
      d0[(size_t)row * NCOLS + colB] = vB;
    } else {
      // col j -> h = j/96, e = (j%96)/3, s = j%3  (s: 0=q,1=k,2=v)
      int h = colA / 96, rm = colA % 96, e = rm / 3, s = rm % 3;
      float* dst = (s == 0) ? d0 : (s == 1) ? d1 : d2;
      dst[((size_t)h * NMTOK + row) * EDIM + e] = vA;
      h = colB / 96; rm = colB % 96; e = rm / 3; s = rm % 3;
      dst = (s == 0) ? d0 : (s == 1) ? d1 : d2;
      dst[((size_t)h * NMTOK + row) * EDIM + e] = vB;
    }
  }
}

// ================= Kernel 3: per-ball key means =================
__global__ __launch_bounds__(256) void kmean_kernel(
    const float* __restrict__ kbuf, float* __restrict__ km) {
  int o = blockIdx.x * 256 + threadIdx.x;
  int e = o & 31;
  int b = (o >> 5) & 511;
  int h = o >> 14;
  const float* p = kbuf + ((size_t)h * NMTOK + b * 16) * EDIM + e;
  float s = 0.f;
#pragma unroll
  for (int m = 0; m < 16; ++m) s += p[m * EDIM];
  km[o] = s * (1.f / 16.f);
}

// ================= Kernel 4: fused sim + top-2 (kmean head in LDS) =================
__global__ __launch_bounds__(256) void simtopk_kernel(
    const float* __restrict__ qbuf, const float* __restrict__ km,
    int* __restrict__ idx) {
  __shared__ float skm[NBALL * EDIM];   // 64 KB
  const int h = blockIdx.y;
  const int t = threadIdx.x;
#pragma unroll
  for (int ii = 0; ii < 16; ++ii) {
    int f4 = t + 256 * ii;
    *(float4*)&skm[f4 * 4] = *(const float4*)&km[(size_t)h * NBALL * EDIM + f4 * 4];
  }
  __syncthreads();

  const int q = blockIdx.x * 256 + t;
  const float* qp = qbuf + ((size_t)h * NMTOK + q) * EDIM;
  float qv[EDIM];
#pragma unroll
  for (int e = 0; e < EDIM; e += 4) {
    float4 xx = *(const float4*)&qp[e];
    qv[e] = xx.x; qv[e + 1] = xx.y; qv[e + 2] = xx.z; qv[e + 3] = xx.w;
  }
  float v1 = -3.4e38f, v2 = -3.4e38f;
  int i1 = 0, i2 = 0;
  for (int b = 0; b < NBALL; ++b) {
    const float* kp = &skm[b * EDIM];
    float d = 0.f;
#pragma unroll
    for (int e = 0; e < EDIM; e += 4) {
      float4 kk = *(const float4*)&kp[e];           // LDS broadcast across lanes
      d += qv[e] * kk.x + qv[e + 1] * kk.y + qv[e + 2] * kk.z + qv[e + 3] * kk.w;
    }
    if (d > v1) { v2 = v1; i2 = i1; v1 = d; i1 = b; }
    else if (d > v2) { v2 = d; i2 = b; }
  }
  int* op = idx + ((size_t)h * NMTOK + q) * 2;
  op[0] = i1; op[1] = i2;
}

// ================= Kernel 5: gathered 32-key softmax attention =================
__global__ __launch_bounds__(256) void attn_kernel(
    const float* __restrict__ qbuf, const float* __restrict__ kbuf,
    const float* __restrict__ vbuf, const int* __restrict__ idx,
    float* __restrict__ ao) {
  int f = blockIdx.x * 256 + threadIdx.x;
  int q = f & (NMTOK - 1);
  int h = f >> 13;
  const float* qp = qbuf + ((size_t)h * NMTOK + q) * EDIM;
  float qv[EDIM];
#pragma unroll
  for (int e = 0; e < EDIM; e += 4) {
    float4 xx = *(const float4*)&qp[e];
    qv[e] = xx.x; qv[e + 1] = xx.y; qv[e + 2] = xx.z; qv[e + 3] = xx.w;
  }
  const int* ip = idx + ((size_t)h * NMTOK + q) * 2;
  int b0 = ip[0], b1 = ip[1];

  float lg[32];
  float mx = -3.4e38f;
#pragma unroll
  for (int kk = 0; kk < 32; ++kk) {
    int ball = (kk < 16) ? b0 : b1;
    int m = kk & 15;
    const float* kp = kbuf + ((size_t)h * NMTOK + ball * 16 + m) * EDIM;
    float d = 0.f;
#pragma unroll
    for (int e = 0; e < EDIM; e += 4) {
      float4 kv = *(const float4*)&kp[e];
      d += qv[e] * kv.x + qv[e + 1] * kv.y + qv[e + 2] * kv.z + qv[e + 3] * kv.w;
    }
    lg[kk] = d * 0.17677669529663687f;   // 1/sqrt(32)
    mx = fmaxf(mx, lg[kk]);
  }
  float se = 0.f;
#pragma unroll
  for (int kk = 0; kk < 32; ++kk) { lg[kk] = __expf(lg[kk] - mx); se += lg[kk]; }
  float inv = 1.f / se;

  float acc[EDIM];
#pragma unroll
  for (int e = 0; e < EDIM; ++e) acc[e] = 0.f;
#pragma unroll
  for (int kk = 0; kk < 32; ++kk) {
    int ball = (kk < 16) ? b0 : b1;
    int m = kk & 15;
    const float* vp = vbuf + ((size_t)h * NMTOK + ball * 16 + m) * EDIM;
    float w = lg[kk] * inv;
#pragma unroll
    for (int e = 0; e < EDIM; e += 4) {
      float4 vv = *(const float4*)&vp[e];
      acc[e] += w * vv.x; acc[e + 1] += w * vv.y;
      acc[e + 2] += w * vv.z; acc[e + 3] += w * vv.w;
    }
  }
  float* op = ao + (size_t)q * DIM + h * EDIM;   // [nm, H*E]
#pragma unroll
  for (int e = 0; e < EDIM; e += 4) {
    float4 vv = { acc[e], acc[e + 1], acc[e + 2], acc[e + 3] };
    *(float4*)&op[e] = vv;
  }
}

// ================= launch =================
extern "C" void kernel_launch(void* const* d_in, const int* in_sizes, int n_in,
                              void* d_out, int out_size, void* d_ws, size_t ws_size,
                              hipStream_t stream) {
  (void)in_sizes; (void)n_in; (void)out_size; (void)ws_size;
  const float* x      = (const float*)d_in[0];
  const float* pos    = (const float*)d_in[1];
  const float* W_qkv  = (const float*)d_in[2];
  const float* b_qkv  = (const float*)d_in[3];
  const float* W_pe   = (const float*)d_in[4];
  const float* b_pe   = (const float*)d_in[5];
  const float* W_proj = (const float*)d_in[6];
  const float* b_proj = (const float*)d_in[7];
  float* out = (float*)d_out;

  float* ws   = (float*)d_ws;
  float* xpe  = ws + XPE_OFF;
  float* qb   = ws + Q_OFF;
  float* kb   = ws + K_OFF;
  float* vb   = ws + V_OFF;
  float* ao   = ws + AO_OFF;
  float* km   = ws + KM_OFF;
  int*   idx  = (int*)(ws + IDX_OFF);

  pe_kernel<<<NBALL, 256, 0, stream>>>(x, pos, W_pe, b_pe, xpe);

  gemm_wmma_f32<0, 768><<<dim3(768 / 64, NMTOK / 64), 256, 0, stream>>>(
      xpe, W_qkv, b_qkv, qb, kb, vb);

  kmean_kernel<<<(NH * NBALL * EDIM) / 256, 256, 0, stream>>>(kb, km);

  simtopk_kernel<<<dim3(NMTOK / 256, NH), 256, 0, stream>>>(qb, km, idx);

  attn_kernel<<<(NH * NMTOK) / 256, 256, 0, stream>>>(qb, kb, vb, idx, ao);

  gemm_wmma_f32<1, 256><<<dim3(256 / 64, NMTOK / 64), 256, 0, stream>>>(
      ao, W_proj, b_proj, out, nullptr, nullptr);
}